// MatrixNetwork_80169859547517
// MI455X (gfx1250) — compile-verified
//
#include <hip/hip_runtime.h>
#include <hip/hip_bf16.h>

// ---------------------------------------------------------------------------
// Sequential chain of 512 f32 1024x1024 GEMMs on the CDNA5 matrix pipe
// (V_WMMA_F32_16X16X4_F32) with Tensor-Data-Mover (TDM) staged, double-
// buffered LDS tiles, then matvec + 13-way projection.
// ---------------------------------------------------------------------------

typedef __attribute__((ext_vector_type(2))) float v2f;
typedef __attribute__((ext_vector_type(4))) float v4f;
typedef __attribute__((ext_vector_type(8))) float v8f;
typedef __attribute__((ext_vector_type(4))) unsigned int u32x4;
typedef __attribute__((ext_vector_type(4))) int i32x4;
typedef __attribute__((ext_vector_type(8))) int i32x8;

#ifndef __has_builtin
#define __has_builtin(x) 0
#endif
#if __has_builtin(__builtin_amdgcn_tensor_load_to_lds) && \
    __has_builtin(__builtin_amdgcn_s_wait_tensorcnt)
#define USE_TDM 1
#else
#define USE_TDM 0
#endif

#define NMAT 1024
#define TM   128      // workgroup tile M
#define TN   128      // workgroup tile N
#define TK   32       // K chunk staged through LDS
#define LDA  36       // A-tile LDS row stride (floats): 36*m % 64 distinct, m=0..15
#define LDB  136      // B-tile LDS row stride: 2*136 % 64 == 16 -> half-wave bank split
#define NCHUNK (NMAT / TK)

#if USE_TDM
// LDS byte offset of a __shared__ object (generic LDS addr low 32 bits).
__device__ __forceinline__ unsigned lds_off(const void* p) {
  return (unsigned)(unsigned long long)p;
}

// D# group 0: count=1, lds addr, 57-bit global addr, type=2 ("image").
__device__ __forceinline__ u32x4 tdm_g0(unsigned lds, const void* gptr) {
  const unsigned long long ga = (unsigned long long)gptr;
  u32x4 g;
  g[0] = 1u;                                            // count=1, no gather
  g[1] = lds;                                           // lds_addr
  g[2] = (unsigned)ga;                                  // global_addr[31:0]
  g[3] = ((unsigned)(ga >> 32) & 0x01FFFFFFu) | (2u << 30); // [56:32] | type=2
  return g;
}

// D# group 1 for a 2-D f32 tile of a 1024x1024 row-major tensor with LDS
// row padding: rows of `tile0` DWORDs, pad after each row.
__device__ __forceinline__ i32x8 tdm_g1(unsigned pad_interval, unsigned pad_amount,
                                        unsigned tile0, unsigned tile1) {
  i32x8 g;
  g[0] = (int)((2u << 16)                 // data_size = 4 bytes
             | (1u << 20)                 // pad_enable
             | (pad_interval << 22)       // 2^(v+1) DWORDs between pads
             | (pad_amount << 25));       // (n+1) DWORDs of pad -> n
  g[1] = (int)((unsigned)(NMAT & 0xFFFF) << 16);   // tensor_dim0[15:0]
  g[2] = (int)((unsigned)(NMAT & 0xFFFF) << 16);   // tdim0 hi=0 | tensor_dim1[15:0]
  g[3] = (int)((tile0 & 0xFFFFu) << 16);           // tdim1 hi=0 | tile_dim0
  g[4] = (int)(tile1 & 0xFFFFu);                   // tile_dim1 | tile_dim2=0
  g[5] = NMAT;                                     // tensor_dim0_stride[31:0]
  g[6] = 0;                                        // stride0 hi | stride1 lo (2-D: unused)
  g[7] = 0;
  return g;
}

__device__ __forceinline__ void tdm_issue(u32x4 g0, i32x8 g1) {
  i32x4 z4 = (i32x4)0;
#if defined(__clang_major__) && (__clang_major__ >= 23)
  i32x8 z8 = (i32x8)0;
  __builtin_amdgcn_tensor_load_to_lds(g0, g1, z4, z4, z8, 0);
#else
  __builtin_amdgcn_tensor_load_to_lds(g0, g1, z4, z4, 0);
#endif
}
#endif  // USE_TDM

// One chain step: Pout = P @ token_mats[token_ids[step]]
__global__ __launch_bounds__(256) void gemm_step_kernel(
    const float* __restrict__ P,
    const float* __restrict__ token_mats,
    const int*   __restrict__ token_ids,
    int step,
    float* __restrict__ Pout)
{
  __shared__ float As[2][TM * LDA];   // 2 x 18.0 KB
  __shared__ float Bs[2][TK * LDB];   // 2 x 17.0 KB

  const int tok = token_ids[step];
  const float* __restrict__ M = token_mats + (size_t)tok * NMAT * NMAT;

  const int bx = blockIdx.x & (NMAT / TN - 1);   // N block (8)
  const int by = blockIdx.x >> 3;                // M block (8)
  const int n0 = bx * TN;
  const int m0 = by * TM;

  const int t    = threadIdx.x;
  const int lane = t & 31;
  const int wave = t >> 5;       // 0..7
  const int half = lane >> 4;    // 0: lanes 0-15, 1: lanes 16-31
  const int l16  = lane & 15;

  // 8 waves arranged 2(M) x 4(N); each wave computes 64(M) x 32(N)
  const int wm0 = (wave & 1) * 64;
  const int wn0 = (wave >> 1) * 32;

  v8f acc[4][2];
#pragma unroll
  for (int mi = 0; mi < 4; ++mi)
#pragma unroll
    for (int ni = 0; ni < 2; ++ni)
      acc[mi][ni] = (v8f)(0.0f);

#if USE_TDM
  const i32x8 g1A = tdm_g1(/*interval 32 DW*/4, /*pad 4 DW*/3, TK, TM);
  const i32x8 g1B = tdm_g1(/*interval 128 DW*/6, /*pad 8 DW*/7, TN, TK);
  const float* Abase = P + (size_t)m0 * NMAT;   // + k0 per chunk
  const float* Bbase = M + n0;                  // + k0*NMAT per chunk

  // Prologue: DMA chunk 0 into buffer 0 (wave 0 only; EXEC ignored by TDM).
  if (t < 32) {
    tdm_issue(tdm_g0(lds_off(&As[0][0]), Abase), g1A);
    tdm_issue(tdm_g0(lds_off(&Bs[0][0]), Bbase), g1B);
  }
#endif

  for (int ch = 0; ch < NCHUNK; ++ch) {
    const int k0  = ch * TK;
    const int cur = ch & 1;

#if USE_TDM
    if (t < 32) {
      if (ch + 1 < NCHUNK) {
        // Kick next chunk's DMA into the other buffer, then retire current
        // chunk's pair (<=2 outstanding keeps the new pair in flight).
        const int nk = k0 + TK;
        tdm_issue(tdm_g0(lds_off(&As[cur ^ 1][0]), Abase + nk), g1A);
        tdm_issue(tdm_g0(lds_off(&Bs[cur ^ 1][0]), Bbase + (size_t)nk * NMAT), g1B);
        __builtin_amdgcn_s_wait_tensorcnt((short)2);
      } else {
        __builtin_amdgcn_s_wait_tensorcnt((short)0);
      }
    }
    __syncthreads();   // publish DMA'd tiles to all waves
#else
    __syncthreads();   // previous compute done before overwrite
    // ---- Stage A tile: 128 rows x 32 cols. 1024 float4 / 256 threads.
#pragma unroll
    for (int i = 0; i < 4; ++i) {
      const int idx  = t + i * 256;
      const int row  = idx >> 3;
      const int col4 = idx & 7;
      const v4f val = *(const v4f*)(P + (size_t)(m0 + row) * NMAT + k0 + col4 * 4);
      *(v4f*)(&As[cur][row * LDA + col4 * 4]) = val;
    }
    // ---- Stage B tile: 32 rows x 128 cols.
#pragma unroll
    for (int i = 0; i < 4; ++i) {
      const int idx  = t + i * 256;
      const int row  = idx >> 5;
      const int col4 = idx & 31;
      const v4f val = *(const v4f*)(M + (size_t)(k0 + row) * NMAT + n0 + col4 * 4);
      *(v4f*)(&Bs[cur][row * LDB + col4 * 4]) = val;
    }
    __syncthreads();
#endif

    const float* Asc = &As[cur][0];
    const float* Bsc = &Bs[cur][0];

    // ---- Consume the chunk: 8 k-steps of 4, 8 WMMAs each.
#pragma unroll
    for (int kk = 0; kk < TK; kk += 4) {
      // A frag per lane: (A[m][kb], A[m][kb+1]), kb = kk + 2*half
      v2f a[4];
#pragma unroll
      for (int mi = 0; mi < 4; ++mi) {
        const int row = wm0 + mi * 16 + l16;
        a[mi] = *(const v2f*)(&Asc[row * LDA + kk + 2 * half]);
      }
      // B frag per lane: (B[kb][n], B[kb+1][n])
      v2f b[2];
#pragma unroll
      for (int ni = 0; ni < 2; ++ni) {
        const int n  = wn0 + ni * 16 + l16;
        const int r0 = (kk + 2 * half) * LDB + n;
        v2f tmp = { Bsc[r0], Bsc[r0 + LDB] };
        b[ni] = tmp;
      }
#pragma unroll
      for (int mi = 0; mi < 4; ++mi)
#pragma unroll
        for (int ni = 0; ni < 2; ++ni)
          acc[mi][ni] = __builtin_amdgcn_wmma_f32_16x16x4_f32(
              false, a[mi], false, b[ni], (short)0, acc[mi][ni], false, false);
    }

#if USE_TDM
    __syncthreads();   // all waves done reading `cur` before it is re-DMA'd
#endif
  }

  // ---- Epilogue: D layout = VGPR r -> row (r + 8*half), col l16
#pragma unroll
  for (int mi = 0; mi < 4; ++mi) {
#pragma unroll
    for (int ni = 0; ni < 2; ++ni) {
      const int row0 = m0 + wm0 + mi * 16 + half * 8;
      const int col  = n0 + wn0 + ni * 16 + l16;
#pragma unroll
      for (int r = 0; r < 8; ++r)
        Pout[(size_t)(row0 + r) * NMAT + col] = acc[mi][ni][r];
    }
  }
}

// v[row] = dot(P[row,:], q)
__global__ __launch_bounds__(256) void matvec_kernel(
    const float* __restrict__ P, const float* __restrict__ q, float* __restrict__ v)
{
  __shared__ float red[256];
  const int row = blockIdx.x;
  float s = 0.0f;
  for (int j = threadIdx.x; j < NMAT; j += 256)
    s += P[(size_t)row * NMAT + j] * q[j];
  red[threadIdx.x] = s;
  __syncthreads();
  for (int off = 128; off > 0; off >>= 1) {
    if (threadIdx.x < off) red[threadIdx.x] += red[threadIdx.x + off];
    __syncthreads();
  }
  if (threadIdx.x == 0) v[row] = red[0];
}

// logits[k] = dot(decode_vecs[k,:], v)
__global__ __launch_bounds__(256) void logits_kernel(
    const float* __restrict__ dv, const float* __restrict__ v, float* __restrict__ out)
{
  __shared__ float red[256];
  const int k = blockIdx.x;
  float s = 0.0f;
  for (int j = threadIdx.x; j < NMAT; j += 256)
    s += dv[(size_t)k * NMAT + j] * v[j];
  red[threadIdx.x] = s;
  __syncthreads();
  for (int off = 128; off > 0; off >>= 1) {
    if (threadIdx.x < off) red[threadIdx.x] += red[threadIdx.x + off];
    __syncthreads();
  }
  if (threadIdx.x == 0) out[k] = red[0];
}

extern "C" void kernel_launch(void* const* d_in, const int* in_sizes, int n_in,
                              void* d_out, int out_size, void* d_ws, size_t ws_size,
                              hipStream_t stream) {
  // setup_inputs() order: token_ids, base_mat, token_mats, decode_vecs, query
  const int*   token_ids   = (const int*)  d_in[0];
  const float* base_mat    = (const float*)d_in[1];
  const float* token_mats  = (const float*)d_in[2];
  const float* decode_vecs = (const float*)d_in[3];
  const float* query       = (const float*)d_in[4];
  float*       out         = (float*)d_out;

  const int seq_len = in_sizes[0];           // 512
  const int vocab   = out_size;              // 13

  float* bufA = (float*)d_ws;                // P ping
  float* bufB = bufA + (size_t)NMAT * NMAT;  // P pong
  float* vvec = bufB + (size_t)NMAT * NMAT;  // v (1024 floats)

  // Seed the chain: P0 = base_mat (capture-safe async d2d copy).
  hipMemcpyAsync(bufA, base_mat, (size_t)NMAT * NMAT * sizeof(float),
                 hipMemcpyDeviceToDevice, stream);

  const dim3 block(256);
  const dim3 grid((NMAT / TM) * (NMAT / TN));   // 64 workgroups of 8 waves

  float* src = bufA;
  float* dst = bufB;
  for (int s = 0; s < seq_len; ++s) {
    gemm_step_kernel<<<grid, block, 0, stream>>>(src, token_mats, token_ids, s, dst);
    float* tmp = src; src = dst; dst = tmp;
  }

  matvec_kernel<<<dim3(NMAT), block, 0, stream>>>(src, query, vvec);
  logits_kernel<<<dim3(vocab), block, 0, stream>>>(decode_vecs, vvec, out);
}